// NormalizedAsymmetricCPGController_17128329576476
// MI455X (gfx1250) — compile-verified
//
#include <hip/hip_runtime.h>
#include <math.h>

typedef float v2f __attribute__((ext_vector_type(2)));
typedef float v8f __attribute__((ext_vector_type(8)));

#define ALPHA_C 0.1f
#define DT_C    0.01f
#define DIFF_C  10.0f
#define EPS_C   1e-9f
#define HB_C    1.5707963267948966f

// One workgroup does everything:
//  phase 1 (parallel): per-row constants into LDS + partial sums of
//    Cx=sum(c*x), Sx=sum(s*x), Cy=sum(c*y), Sy=sum(s*y)
//  phase 2 (wave 0): WMMA ones-matmul reduction of the 256 partials/quantity
//  phase 3 (lane 0): O(N) sequential recurrence on P=Cx+Sy, Q=Sx-Cy
__global__ __launch_bounds__(256)
void cpg_kernel(const float* __restrict__ phase,
                const float* __restrict__ amplitudes,
                const float* __restrict__ w,
                const float* __restrict__ ha,
                const float* __restrict__ bcoef,
                const float* __restrict__ xy,
                const float* __restrict__ xy_dot_old,
                float* __restrict__ out,
                int n) {
    extern __shared__ char smem_raw[];
    float4* recA = (float4*)(smem_raw);                       // ci, si, K1, K2
    float4* recB = (float4*)(smem_raw + (size_t)n * 16);      // lo1, hi1, lo2, hi2
    float2* recC = (float2*)(smem_raw + (size_t)n * 32);      // A0, A1
    float*  part = (float*) (smem_raw + (size_t)n * 40);      // 4 * 256 partials

    const int tid = threadIdx.x;

    // ---------------- Phase 1: parallel precompute ----------------
    float pcx = 0.0f, psx = 0.0f, pcy = 0.0f, psy = 0.0f;
    for (int j = tid; j < n; j += 256) {
        float p  = phase[j];
        float cj = cosf(p);
        float sj = sinf(p);
        float x0 = xy[2 * j + 0];
        float y0 = xy[2 * j + 1];
        float r2 = fmaf(x0, x0, y0 * y0);
        float ta = ALPHA_C * (1.0f - r2 * r2);
        float xdo = xy_dot_old[2 * j + 0];
        float ydo = xy_dot_old[2 * j + 1];
        float zeta = 1.0f - ha[j] * ((xdo + EPS_C) / (fabsf(xdo) + EPS_C));
        float tb = w[j] / (zeta + EPS_C);
        // xd = K1 + ci*P + si*Q ; yd = K2 + si*P - ci*Q
        float K1 = ta * x0 - tb * y0 - x0;
        float K2 = tb * x0 + ta * y0 - y0;
        recA[j] = make_float4(cj, sj, K1, K2);
        recB[j] = make_float4(xdo - DIFF_C, xdo + DIFF_C, ydo - DIFF_C, ydo + DIFF_C);
        // angle_i = clip(A0 + A1*yd)  with  A0 = amp*y0 + b, A1 = amp*DT
        recC[j] = make_float2(fmaf(amplitudes[j], y0, bcoef[j]), amplitudes[j] * DT_C);
        pcx = fmaf(cj, x0, pcx);
        psx = fmaf(sj, x0, psx);
        pcy = fmaf(cj, y0, pcy);
        psy = fmaf(sj, y0, psy);
    }
    part[0 * 256 + tid] = pcx;
    part[1 * 256 + tid] = psx;
    part[2 * 256 + tid] = pcy;
    part[3 * 256 + tid] = psy;
    __syncthreads();

    // ---------------- Phase 2: WMMA reduction (wave 0, EXEC all ones) ------
    if (tid < 32) {
        const int lane = tid;
        v2f ones;
        ones.x = 1.0f;
        ones.y = 1.0f;
        float tot0 = 0.0f, tot1 = 0.0f, tot2 = 0.0f, tot3 = 0.0f;
        #pragma unroll
        for (int q = 0; q < 4; ++q) {
            v8f acc = {};
            #pragma unroll
            for (int ch = 0; ch < 4; ++ch) {
                // 64 partials per chunk; with B == all-ones every entry of A
                // contributes exactly once to every column of D, so the exact
                // A-operand lane layout does not affect the final total.
                v2f a;
                a.x = part[q * 256 + ch * 64 + lane * 2 + 0];
                a.y = part[q * 256 + ch * 64 + lane * 2 + 1];
                acc = __builtin_amdgcn_wmma_f32_16x16x4_f32(
                    false, a, false, ones, (short)0, acc, false, false);
            }
            // lane L holds D[M=v..v+7 or v+8..v+15, col]; sum 8 VGPRs then
            // combine the two half-wave row groups.
            float s = acc[0] + acc[1] + acc[2] + acc[3] +
                      acc[4] + acc[5] + acc[6] + acc[7];
            s += __shfl_xor(s, 16, 32);
            if (q == 0) tot0 = s;   // Cx
            if (q == 1) tot1 = s;   // Sx
            if (q == 2) tot2 = s;   // Cy
            if (q == 3) tot3 = s;   // Sy
        }

        // ---------------- Phase 3: sequential recurrence (lane 0) ----------
        if (lane == 0) {
            float P = tot0 + tot3;   // Cx + Sy
            float Q = tot1 - tot2;   // Sx - Cy
            // 2-ahead prefetch out of LDS keeps loads off the P/Q chain.
            float4 a0 = recA[0], bb0 = recB[0];
            float2 c0 = recC[0];
            int i1 = (n > 1) ? 1 : 0;
            float4 a1 = recA[i1], bb1 = recB[i1];
            float2 c1 = recC[i1];
            for (int i = 0; i < n; ++i) {
                float4 a = a0;  float4 bb = bb0;  float2 cc = c0;
                a0 = a1; bb0 = bb1; c0 = c1;
                int nx = i + 2;
                if (nx > n - 1) nx = n - 1;
                a1 = recA[nx]; bb1 = recB[nx]; c1 = recC[nx];

                float ci = a.x, si = a.y;
                float xd = fmaf(si, Q, fmaf(ci, P, a.z));
                float yd = fmaf(-ci, Q, fmaf(si, P, a.w));
                xd = fminf(fmaxf(xd, bb.x), bb.y);
                yd = fminf(fmaxf(yd, bb.z), bb.w);
                float dx = xd * DT_C;
                float dy = yd * DT_C;
                P = fmaf(si, dy, fmaf(ci, dx, P));
                Q = fmaf(-ci, dy, fmaf(si, dx, Q));
                float ang = fmaf(cc.y, yd, cc.x);
                ang = fminf(fmaxf(ang, -HB_C), HB_C);
                out[i] = ang;
            }
        }
    }
}

extern "C" void kernel_launch(void* const* d_in, const int* in_sizes, int n_in,
                              void* d_out, int out_size, void* d_ws, size_t ws_size,
                              hipStream_t stream) {
    const float* phase      = (const float*)d_in[0];
    const float* amplitudes = (const float*)d_in[1];
    const float* w          = (const float*)d_in[2];
    const float* ha         = (const float*)d_in[3];
    const float* bcoef      = (const float*)d_in[4];
    const float* xy         = (const float*)d_in[5];
    const float* xy_dot_old = (const float*)d_in[6];
    float* out = (float*)d_out;
    int n = in_sizes[0];

    size_t shmem = (size_t)n * 40 + 4 * 256 * sizeof(float);  // ~164 KB for n=4096
    (void)hipFuncSetAttribute((const void*)cpg_kernel,
                              hipFuncAttributeMaxDynamicSharedMemorySize,
                              (int)shmem);
    cpg_kernel<<<1, 256, shmem, stream>>>(phase, amplitudes, w, ha, bcoef,
                                          xy, xy_dot_old, out, n);
}